// GCN_45655502357027
// MI455X (gfx1250) — compile-verified
//
#include <hip/hip_runtime.h>
#include <math.h>

#define NNODES 50000
#define NEDGES 800000
#define NFEAT  512
#define NHID   128
#define NCLS   40

typedef __attribute__((ext_vector_type(2))) float v2f;
typedef __attribute__((ext_vector_type(8))) float v8f;
typedef __attribute__((ext_vector_type(4))) int   v4i;

// pointer flavors for the async global->LDS builtin (arg0 type taken from the
// round-2 diagnostic: non-const int4 in AS(1); dst mirrored as int4 in AS(3))
typedef __attribute__((address_space(1))) v4i* gpu_v4i_ptr;
typedef __attribute__((address_space(3))) v4i* lds_v4i_ptr;

// ---- gfx1250 async global->LDS path (guarded; falls back to ds_store staging) ----
#if defined(__has_builtin)
#  if __has_builtin(__builtin_amdgcn_global_load_async_to_lds_b128)
#    define USE_ASYNC_LDS 1
#  endif
#endif
#ifndef USE_ASYNC_LDS
#  define USE_ASYNC_LDS 0
#endif

#if USE_ASYNC_LDS
#  if __has_builtin(__builtin_amdgcn_s_wait_asynccnt)
#    define WAIT_ASYNC() __builtin_amdgcn_s_wait_asynccnt(0)
#  else
#    define WAIT_ASYNC() asm volatile("s_wait_asynccnt 0x0" ::: "memory")
#  endif
#endif

// ---------------- degree / normalization ----------------

__global__ void k_fill1(float* __restrict__ p, int n) {
  int i = blockIdx.x * blockDim.x + threadIdx.x;
  if (i < n) p[i] = 1.0f;   // self-loop contributes 1 to every degree
}

__global__ void k_deg_count(const long long* __restrict__ dst, float* __restrict__ deg, int ne) {
  int i = blockIdx.x * blockDim.x + threadIdx.x;
  if (i < ne) atomicAdd(&deg[(int)dst[i]], 1.0f);
}

__global__ void k_rsqrt_inplace(float* __restrict__ p, int n) {
  int i = blockIdx.x * blockDim.x + threadIdx.x;
  if (i < n) {
    float d = p[i];
    p[i] = (d > 0.0f) ? rsqrtf(d) : 0.0f;
  }
}

// ---------------- fp32 WMMA GEMM: C[M,N] = A[M,K] @ B[K,N] ----------------
// Block = 256 threads (8 waves). blockIdx.y selects a 16-column tile; the block
// stages B[:,16] into LDS once (async-to-LDS when available). Each wave owns
// 64 rows (4 x 16-row tiles) of that column tile: one B fragment feeds 4 WMMAs.
//
// V_WMMA_F32_16X16X4_F32 layouts (ISA 7.12.2, 32-bit):
//   A 16x4 : lanes 0-15 hold K=0,1 ; lanes 16-31 hold K=2,3 (per-lane v2f)
//   B 4x16 : VGPR v, lanes 0-15 -> K=v ; lanes 16-31 -> K=v+2
//   C/D    : VGPR r, lanes 0-15 -> (M=r, N=lane) ; lanes 16-31 -> (M=r+8, N=lane-16)
template<int K, int LDA, int LDB, int LDC, int NVALID>
__global__ __launch_bounds__(256) void k_gemm_wmma_f32(const float* __restrict__ A,
                                                       const float* __restrict__ B,
                                                       float* __restrict__ C, int M) {
  __shared__ float Bs[K * 16];               // [k][n] contiguous, 16 floats per row
  const int tid  = threadIdx.x;
  const int col0 = blockIdx.y << 4;

  // ---- cooperative stage of B[:, col0:col0+16] ----
  const float* Bg = B + col0;
  constexpr int CHUNKS = (K * 16) / 4;       // float4 chunks
#pragma unroll
  for (int i = 0; i < CHUNKS / 256; ++i) {
    int c = tid + i * 256;
    int k = c >> 2;
    int n = (c & 3) << 2;
#if USE_ASYNC_LDS
    __builtin_amdgcn_global_load_async_to_lds_b128(
        (gpu_v4i_ptr)(Bg + (size_t)k * LDB + n),
        (lds_v4i_ptr)(Bs + c * 4), 0, 0);
#else
    *(float4*)(Bs + c * 4) = *(const float4*)(Bg + (size_t)k * LDB + n);
#endif
  }
#if USE_ASYNC_LDS
  WAIT_ASYNC();
#endif
  __syncthreads();

  const int wave = tid >> 5;
  const int lane = tid & 31;
  const int half = lane >> 4;                // lane group 0/1
  const int l16  = lane & 15;
  const int rowbase = (blockIdx.x * 8 + wave) * 64;   // 4 row tiles of 16
  if (rowbase >= M) return;                  // wave-uniform; no barriers follow

  // A fragment pointers for the 4 row tiles (clamped in-bounds; stores guarded)
  const float* Ap[4];
#pragma unroll
  for (int t = 0; t < 4; ++t) {
    int r = rowbase + 16 * t + l16;
    if (r >= M) r = M - 1;
    Ap[t] = A + (size_t)r * LDA + (half << 1);
  }
  const float* Bl = Bs + (half << 1) * 16 + l16;   // LDS row (2*half), col l16

  v8f acc0 = {}, acc1 = {}, acc2 = {}, acc3 = {};
  for (int kk = 0; kk < K; kk += 64) {
#pragma unroll
    for (int t = 0; t < 4; ++t)                     // stream-ahead hint for A
      __builtin_prefetch(Ap[t] + kk + 64, 0, 0);
#pragma unroll
    for (int k = kk; k < kk + 64; k += 4) {
      v2f b;
      b.x = Bl[k * 16];                             // B[k + 2*half    ][col]
      b.y = Bl[k * 16 + 16];                        // B[k + 2*half + 1][col]
      v2f a0 = *(const v2f*)(Ap[0] + k);
      v2f a1 = *(const v2f*)(Ap[1] + k);
      v2f a2 = *(const v2f*)(Ap[2] + k);
      v2f a3 = *(const v2f*)(Ap[3] + k);
      acc0 = __builtin_amdgcn_wmma_f32_16x16x4_f32(false, a0, false, b, (short)0, acc0, false, false);
      acc1 = __builtin_amdgcn_wmma_f32_16x16x4_f32(false, a1, false, b, (short)0, acc1, false, false);
      acc2 = __builtin_amdgcn_wmma_f32_16x16x4_f32(false, a2, false, b, (short)0, acc2, false, false);
      acc3 = __builtin_amdgcn_wmma_f32_16x16x4_f32(false, a3, false, b, (short)0, acc3, false, false);
    }
  }

  const int ccol = col0 + l16;
  v8f accs[4] = {acc0, acc1, acc2, acc3};
#pragma unroll
  for (int t = 0; t < 4; ++t) {
    if (rowbase + 16 * t < M && ccol < NVALID) {    // M % 16 == 0 -> full tiles
      float* Cp = C + (size_t)(rowbase + 16 * t + (half << 3)) * LDC + ccol;
#pragma unroll
      for (int r = 0; r < 8; ++r) Cp[(size_t)r * LDC] = accs[t][r];
    }
  }
}

// ---------------- aggregation: self-loop init + edge scatter ----------------

template<int F>
__global__ void k_self_loop_init(const float* __restrict__ h, const float* __restrict__ dinv,
                                 float* __restrict__ out, long long total) {
  long long i = blockIdx.x * (long long)blockDim.x + threadIdx.x;
  if (i >= total) return;
  int node = (int)(i / F);
  float w = dinv[node];
  out[i] = h[i] * w * w;
}

// one thread handles 4 consecutive features of one edge (coalesced rows)
template<int F>
__global__ void k_agg_edges(const float* __restrict__ h, const float* __restrict__ dinv,
                            const long long* __restrict__ src, const long long* __restrict__ dst,
                            float* __restrict__ out, long long ne) {
  constexpr int F4 = F / 4;
  long long idx = blockIdx.x * (long long)blockDim.x + threadIdx.x;
  long long e = idx / F4;
  if (e >= ne) return;
  int g = (int)(idx - e * F4);
  int s = (int)src[e];
  int d = (int)dst[e];
  float norm = dinv[s] * dinv[d];
  const float4 v = *(const float4*)(h + (size_t)s * F + (g << 2));
  float* o = out + (size_t)d * F + (g << 2);
  atomicAdd(o + 0, v.x * norm);
  atomicAdd(o + 1, v.y * norm);
  atomicAdd(o + 2, v.z * norm);
  atomicAdd(o + 3, v.w * norm);
}

template<int F>
__global__ void k_bias_relu(float* __restrict__ x, const float* __restrict__ b,
                            long long total) {
  long long i = blockIdx.x * (long long)blockDim.x + threadIdx.x;
  if (i >= total) return;
  int f = (int)(i % F);
  float v = x[i] + b[f];
  x[i] = v > 0.0f ? v : 0.0f;
}

// ---------------- bias + log-softmax over NCLS=40 (in place on d_out) ----------------

__global__ void k_bias_logsoftmax(float* __restrict__ out, const float* __restrict__ b, int n) {
  int i = blockIdx.x * blockDim.x + threadIdx.x;
  if (i >= n) return;
  float* row = out + (size_t)i * NCLS;
  float z[NCLS];
  float m = -INFINITY;
#pragma unroll
  for (int c = 0; c < NCLS; ++c) {
    z[c] = row[c] + b[c];
    m = fmaxf(m, z[c]);
  }
  float s = 0.0f;
#pragma unroll
  for (int c = 0; c < NCLS; ++c) s += __expf(z[c] - m);
  float ls = __logf(s);
#pragma unroll
  for (int c = 0; c < NCLS; ++c) row[c] = z[c] - m - ls;
}

// ---------------- host launch ----------------

extern "C" void kernel_launch(void* const* d_in, const int* in_sizes, int n_in,
                              void* d_out, int out_size, void* d_ws, size_t ws_size,
                              hipStream_t stream) {
  const float*      x   = (const float*)d_in[0];
  const long long*  ei  = (const long long*)d_in[1];   // int64 edge_index [2, E]
  const float*      W1  = (const float*)d_in[2];
  const float*      b1  = (const float*)d_in[3];
  const float*      W2  = (const float*)d_in[4];
  const float*      b2  = (const float*)d_in[5];
  float*            out = (float*)d_out;               // [N, 40] float32

  const long long* src = ei;
  const long long* dst = ei + NEDGES;

  char* ws = (char*)d_ws;
  float* dinv = (float*)ws;                                            // N floats
  float* h1   = (float*)(ws + (((size_t)NNODES * 4 + 255) & ~255ull)); // N*128
  float* agg1 = h1 + (size_t)NNODES * NHID;                            // N*128
  float* h2   = h1;   // h1 region is dead once agg1+relu are built; reuse for N*40

  const int T = 256;
  const int ROWS_PER_BLK = 8 * 64;                                     // 8 waves x 64 rows
  const int rowblks = (NNODES + ROWS_PER_BLK - 1) / ROWS_PER_BLK;      // 98

  // degrees -> D^-1/2
  k_fill1<<<(NNODES + T - 1) / T, T, 0, stream>>>(dinv, NNODES);
  k_deg_count<<<(NEDGES + T - 1) / T, T, 0, stream>>>(dst, dinv, NEDGES);
  k_rsqrt_inplace<<<(NNODES + T - 1) / T, T, 0, stream>>>(dinv, NNODES);

  // ---- layer 1: h1 = x @ W1 ----
  k_gemm_wmma_f32<NFEAT, NFEAT, NHID, NHID, NHID>
      <<<dim3(rowblks, NHID / 16), T, 0, stream>>>(x, W1, h1, NNODES);
  {
    long long total = (long long)NNODES * NHID;
    k_self_loop_init<NHID><<<(int)((total + T - 1) / T), T, 0, stream>>>(h1, dinv, agg1, total);
    long long eth = (long long)NEDGES * (NHID / 4);
    k_agg_edges<NHID><<<(int)((eth + T - 1) / T), T, 0, stream>>>(h1, dinv, src, dst, agg1, NEDGES);
    k_bias_relu<NHID><<<(int)((total + T - 1) / T), T, 0, stream>>>(agg1, b1, total);
  }

  // ---- layer 2: h2 = relu(agg1) @ W2 ----
  k_gemm_wmma_f32<NHID, NHID, NCLS, NCLS, NCLS>
      <<<dim3(rowblks, (NCLS + 15) / 16), T, 0, stream>>>(agg1, W2, h2, NNODES);
  {
    long long total = (long long)NNODES * NCLS;
    // aggregate layer 2 directly into d_out (fully overwritten, then atomically updated)
    k_self_loop_init<NCLS><<<(int)((total + T - 1) / T), T, 0, stream>>>(h2, dinv, out, total);
    long long eth = (long long)NEDGES * (NCLS / 4);
    k_agg_edges<NCLS><<<(int)((eth + T - 1) / T), T, 0, stream>>>(h2, dinv, src, dst, out, NEDGES);
    k_bias_logsoftmax<<<(NNODES + T - 1) / T, T, 0, stream>>>(out, b2, NNODES);
  }
}